// KNNReg_39135742001605
// MI455X (gfx1250) — compile-verified
//
#include <hip/hip_runtime.h>
#include <math.h>

typedef __attribute__((ext_vector_type(2))) float v2f;
typedef __attribute__((ext_vector_type(8))) float v8f;

#define BATCH  2
#define NPTS   8192
#define KNN_K  3
#define NTILES (NPTS / 16)

// Branchless sorted-insert of (d, j) into min-list t0 <= t1 <= t2.
// Compiles to v_cmp + v_cndmask only (no EXEC-mask branching).
__device__ __forceinline__ void ins3(float d, int j,
                                     float& t0, float& t1, float& t2,
                                     int& j0, int& j1, int& j2) {
  const bool c2 = d < t2;
  const bool c1 = d < t1;
  const bool c0 = d < t0;
  float nt2 = c1 ? t1 : (c2 ? d : t2);
  int   nj2 = c1 ? j1 : (c2 ? j : j2);
  float nt1 = c0 ? t0 : (c1 ? d : t1);
  int   nj1 = c0 ? j0 : (c1 ? j : j1);
  float nt0 = c0 ? d : t0;
  int   nj0 = c0 ? j : j0;
  t0 = nt0; t1 = nt1; t2 = nt2;
  j0 = nj0; j1 = nj1; j2 = nj2;
}

__global__ void init_out_kernel(float* out) {
  if (threadIdx.x == 0 && blockIdx.x == 0) out[0] = 0.0f;
}

// One wave (32 threads) per 16-row tile. Gram tile via V_WMMA_F32_16X16X4_F32.
__global__ __launch_bounds__(32)
void knn_wmma_kernel(const float* __restrict__ xyz,
                     int* __restrict__ nn_ind,
                     float* __restrict__ out) {
  const int tile = blockIdx.x;                 // 0 .. BATCH*NTILES-1
  const int b    = tile / NTILES;
  const int i0   = (tile - b * NTILES) * 16;   // first row of this tile
  const int lane = threadIdx.x;                // wave32
  const int half = lane >> 4;                  // 0: K=0..1, 1: K=2..3
  const int lm   = lane & 15;
  const float* X = xyz + (size_t)b * NPTS * 3;

  // A fragment: 16x4 f32, A[m][k] = xyz[i0+m][k], k=3 padded with 0.
  // ISA layout: lanes 0-15 hold (K=0,K=1) of row lm; lanes 16-31 hold (K=2,K=3).
  const float* pr = X + (size_t)(i0 + lm) * 3;
  float rx = pr[0], ry = pr[1], rz = pr[2];
  v2f a;
  a.x = half ? rz   : rx;
  a.y = half ? 0.0f : ry;

  // Row squared norms: computed fully in-lane, broadcast ONCE outside the loop.
  float fullsq = rx * rx + ry * ry + rz * rz;  // sq[i0+lm] on every lane
  float rs[8];
#pragma unroll
  for (int r = 0; r < 8; ++r) rs[r] = __shfl(fullsq, r + 8 * half, 32);

  // Running top-3 per (row r, column-slot lm).
  float t0a[8], t1a[8], t2a[8];
  int   i0a[8], i1a[8], i2a[8];
#pragma unroll
  for (int r = 0; r < 8; ++r) {
    t0a[r] = t1a[r] = t2a[r] = 3.0e38f;
    i0a[r] = i1a[r] = i2a[r] = 0;
  }

#pragma unroll 2
  for (int j0 = 0; j0 < NPTS; j0 += 16) {
    // B fragment: 4x16 f32, B[k][n] = xyz[j0+n][k] (row-striped across lanes).
    const float* pc = X + (size_t)(j0 + lm) * 3;
    float cx = pc[0], cy = pc[1], cz = pc[2];
    v2f bb;
    bb.x = half ? cz   : cx;
    bb.y = half ? 0.0f : cy;

    // Column norm fully in-lane: no DS permute / dscnt stall in the hot loop.
    float sqj = cx * cx + cy * cy + cz * cz;   // sq[j0+lm] (column n = lm)

    v8f c = {};
    c = __builtin_amdgcn_wmma_f32_16x16x4_f32(
        /*neg_a=*/false, a, /*neg_b=*/false, bb,
        /*c_mod=*/(short)0, c, /*reuse_a=*/false, /*reuse_b=*/false);

    const int jidx = j0 + lm;
#pragma unroll
    for (int r = 0; r < 8; ++r) {
      // D[m][n] for m = r + 8*half, n = lm
      float d = rs[r] + sqj - 2.0f * c[r];
      ins3(d, jidx, t0a[r], t1a[r], t2a[r], i0a[r], i1a[r], i2a[r]);
    }
  }

  // Butterfly merge across the 16 column slots (stays inside each half-wave).
#pragma unroll
  for (int off = 1; off < 16; off <<= 1) {
#pragma unroll
    for (int r = 0; r < 8; ++r) {
      float o0 = __shfl_xor(t0a[r], off, 32);
      float o1 = __shfl_xor(t1a[r], off, 32);
      float o2 = __shfl_xor(t2a[r], off, 32);
      int   p0 = __shfl_xor(i0a[r], off, 32);
      int   p1 = __shfl_xor(i1a[r], off, 32);
      int   p2 = __shfl_xor(i2a[r], off, 32);
      ins3(o0, p0, t0a[r], t1a[r], t2a[r], i0a[r], i1a[r], i2a[r]);
      ins3(o1, p1, t0a[r], t1a[r], t2a[r], i0a[r], i1a[r], i2a[r]);
      ins3(o2, p2, t0a[r], t1a[r], t2a[r], i0a[r], i1a[r], i2a[r]);
    }
  }

  if (lm == 0) {  // lane 0 of each half holds rows r + 8*half, r = 0..7
    float dsum = 0.0f;
#pragma unroll
    for (int r = 0; r < 8; ++r) {
      int row = i0 + r + 8 * half;
      size_t base = ((size_t)b * NPTS + row) * KNN_K;
      nn_ind[base + 0] = i0a[r];
      nn_ind[base + 1] = i1a[r];
      nn_ind[base + 2] = i2a[r];
      dsum += t0a[r] + t1a[r] + t2a[r];
    }
    const float scale = 1.0f / ((float)BATCH * (float)NPTS * (float)KNN_K);
    atomicAdd(out, dsum * scale);
  }
}

// Unbiased std of 3 samples (ddof = 1).
__device__ __forceinline__ float std3(float v0, float v1, float v2) {
  float mu = (v0 + v1 + v2) * (1.0f / 3.0f);
  float d0 = v0 - mu, d1 = v1 - mu, d2 = v2 - mu;
  return sqrtf((d0 * d0 + d1 * d1 + d2 * d2) * 0.5f);
}

__global__ __launch_bounds__(256)
void gather_std_kernel(const float* __restrict__ rot,
                       const float* __restrict__ sc,
                       const float* __restrict__ col,
                       const float* __restrict__ op,
                       const int*  __restrict__ nn_ind,
                       float* __restrict__ out) {
  int gid = blockIdx.x * blockDim.x + threadIdx.x;  // (b, i) flattened
  float contrib = 0.0f;
  if (gid < BATCH * NPTS) {
    int b = gid / NPTS;
    size_t nbase = (size_t)gid * KNN_K;
    int n0 = nn_ind[nbase + 0];
    int n1 = nn_ind[nbase + 1];
    int n2 = nn_ind[nbase + 2];
    size_t bn = (size_t)b * NPTS;
    float acc;

    // rotations [B,N,4]
    {
      size_t o0 = (bn + n0) * 4, o1 = (bn + n1) * 4, o2 = (bn + n2) * 4;
      acc = 0.0f;
#pragma unroll
      for (int c = 0; c < 4; ++c) acc += std3(rot[o0 + c], rot[o1 + c], rot[o2 + c]);
      contrib += acc * (1.0f / 4.0f);
    }
    // scales [B,N,3]
    {
      size_t o0 = (bn + n0) * 3, o1 = (bn + n1) * 3, o2 = (bn + n2) * 3;
      acc = 0.0f;
#pragma unroll
      for (int c = 0; c < 3; ++c) acc += std3(sc[o0 + c], sc[o1 + c], sc[o2 + c]);
      contrib += acc * (1.0f / 3.0f);
    }
    // colors [B,N,45]
    {
      size_t o0 = (bn + n0) * 45, o1 = (bn + n1) * 45, o2 = (bn + n2) * 45;
      acc = 0.0f;
      for (int c = 0; c < 45; ++c) acc += std3(col[o0 + c], col[o1 + c], col[o2 + c]);
      contrib += acc * (1.0f / 45.0f);
    }
    // opacity [B,N,1]
    contrib += std3(op[bn + n0], op[bn + n1], op[bn + n2]);

    contrib *= 1.0f / ((float)NPTS * (float)BATCH);
  }

  // wave32 reduction, one atomic per wave
#pragma unroll
  for (int o = 16; o > 0; o >>= 1) contrib += __shfl_down(contrib, o, 32);
  if ((threadIdx.x & 31) == 0) atomicAdd(out, contrib);
}

extern "C" void kernel_launch(void* const* d_in, const int* in_sizes, int n_in,
                              void* d_out, int out_size, void* d_ws, size_t ws_size,
                              hipStream_t stream) {
  const float* xyz = (const float*)d_in[0];  // [B,N,3]
  const float* rot = (const float*)d_in[1];  // [B,N,4]
  const float* sc  = (const float*)d_in[2];  // [B,N,3]
  const float* col = (const float*)d_in[3];  // [B,N,45]
  const float* op  = (const float*)d_in[4];  // [B,N,1]
  float* out = (float*)d_out;
  int*   nn  = (int*)d_ws;                   // [B,N,3] neighbor indices (196 KB)

  init_out_kernel<<<1, 32, 0, stream>>>(out);
  knn_wmma_kernel<<<BATCH * NTILES, 32, 0, stream>>>(xyz, nn, out);
  gather_std_kernel<<<(BATCH * NPTS + 255) / 256, 256, 0, stream>>>(rot, sc, col, op, nn, out);
}